// GAT_66760971649426
// MI455X (gfx1250) — compile-verified
//
#include <hip/hip_runtime.h>
#include <hip/hip_bf16.h>

// Problem constants (match reference setup_inputs)
#define GN   50000     // nodes
#define GE   800000    // edges (dst sorted)
#define GIN  256       // input features
#define GHID 64        // hidden per head
#define GHEADS 4
#define GC   64        // classes per head
#define GSLOPE 0.2f
#define GD   256       // HEADS*HID == HEADS*C == IN == 256 for every GEMM here

typedef __attribute__((ext_vector_type(2))) float v2f;
typedef __attribute__((ext_vector_type(8))) float v8f;

// ---------------------------------------------------------------------------
// CSR row pointers from sorted dst: rowptr[i] = lower_bound(dst, i)
// ---------------------------------------------------------------------------
__global__ void build_rowptr(const int* __restrict__ dst, int* __restrict__ rowptr,
                             int nE, int nN) {
    int i = blockIdx.x * blockDim.x + threadIdx.x;
    if (i > nN) return;
    int lo = 0, hi = nE;
    while (lo < hi) {
        int mid = (lo + hi) >> 1;
        if (dst[mid] < i) lo = mid + 1; else hi = mid;
    }
    rowptr[i] = lo;
}

// ---------------------------------------------------------------------------
// C[N,256] = A[N,256] @ B[256,256], f32 WMMA 16x16x4.
// One wave owns a 16x16 output tile; block = 8 waves = 8 column tiles.
// grid = (N/16, 2). A-frag lanes 0-15: M=lane, K=k0..k0+1; lanes 16-31: K=k0+2..3.
// B-frag lanes 0-15: N=lane, same K split. D: VGPR r -> rows r (lanes 0-15), r+8.
// ---------------------------------------------------------------------------
__global__ void __launch_bounds__(256)
gemm_wmma_f32(const float* __restrict__ A, const float* __restrict__ B,
              float* __restrict__ C) {
    const int K = GD, M = GD;
    const int lane = threadIdx.x & 31;
    const int wave = threadIdx.x >> 5;
    const int m0 = blockIdx.x * 16;
    const int n0 = (blockIdx.y * 8 + wave) * 16;

    const int lrow = lane & 15;        // row within A tile / col within B tile
    const int khalf = (lane >> 4) * 2; // K sub-offset per lane half

    const float* __restrict__ Ap = A + (size_t)(m0 + lrow) * K + khalf;
    const float* __restrict__ Bp = B + (size_t)khalf * M + n0 + lrow;

    v8f c = {};
#pragma unroll 8
    for (int k0 = 0; k0 < K; k0 += 4) {
        v2f a, b;
        a.x = Ap[k0];
        a.y = Ap[k0 + 1];
        b.x = Bp[(size_t)k0 * M];
        b.y = Bp[(size_t)k0 * M + M];
        c = __builtin_amdgcn_wmma_f32_16x16x4_f32(
                /*neg_a=*/false, a, /*neg_b=*/false, b,
                /*c_mod=*/(short)0, c, /*reuse_a=*/false, /*reuse_b=*/false);
    }

    float* __restrict__ Cp = C + (size_t)(m0 + ((lane >> 4) << 3)) * M + n0 + lrow;
#pragma unroll
    for (int r = 0; r < 8; ++r) Cp[(size_t)r * M] = c[r];
}

// ---------------------------------------------------------------------------
// el[n,h] = sum_o h[n,h,o]*al[h,o]; er likewise. One thread per (n,h).
// ---------------------------------------------------------------------------
__global__ void __launch_bounds__(256)
el_er_kernel(const float* __restrict__ h, const float* __restrict__ al,
             const float* __restrict__ ar, float* __restrict__ el,
             float* __restrict__ er) {
    __shared__ float s_al[GD];
    __shared__ float s_ar[GD];
    int tid = threadIdx.x;
    s_al[tid] = al[tid];
    s_ar[tid] = ar[tid];
    __syncthreads();

    int g = blockIdx.x * blockDim.x + tid;
    if (g >= GN * GHEADS) return;
    int n = g >> 2;
    int hd = g & 3;
    const float* __restrict__ hp = h + (size_t)n * GD + hd * GHID;
    float accl = 0.f, accr = 0.f;
#pragma unroll 8
    for (int o = 0; o < GHID; ++o) {
        float v = hp[o];
        accl += v * s_al[hd * GHID + o];
        accr += v * s_ar[hd * GHID + o];
    }
    el[g] = accl;
    er[g] = accr;
}

__device__ __forceinline__ float leaky(float x) {
    return x > 0.f ? x : x * GSLOPE;
}

// ---------------------------------------------------------------------------
// One block (256 thr) per destination node: segment max -> exp-sum -> weighted
// aggregate of h[src] + bias. Thread tid -> (head = tid>>6, chan = tid&63).
// MEAN=true fuses the final head-mean for the last layer.
// ---------------------------------------------------------------------------
template <bool MEAN>
__global__ void __launch_bounds__(256)
gat_aggregate(const float* __restrict__ h, const float* __restrict__ el,
              const float* __restrict__ er, const int* __restrict__ src,
              const int* __restrict__ rowptr, const float* __restrict__ bias,
              float* __restrict__ out) {
    const int d = blockIdx.x;
    const int tid = threadIdx.x;
    const int hd = tid >> 6;
    const int o = tid & 63;

    const int e0 = rowptr[d];
    const int e1 = rowptr[d + 1];

    __shared__ float s_m[GHEADS];
    __shared__ float s_inv[GHEADS];
    __shared__ float s_er[GHEADS];
    __shared__ int   s_src[64];
    __shared__ float s_alpha[64 * GHEADS];
    __shared__ float s_red[256];

    // Per-head max and exp-sum over the incoming-edge segment (4 lanes active;
    // avg degree is 16 so this serial scan is cheap and exactly reference-stable).
    if (tid < GHEADS) {
        float erd = er[(size_t)d * GHEADS + tid];
        s_er[tid] = erd;
        float m = -3.0e38f;
        for (int e = e0; e < e1; ++e)
            m = fmaxf(m, leaky(el[(size_t)src[e] * GHEADS + tid] + erd));
        float s = 0.f;
        for (int e = e0; e < e1; ++e)
            s += __expf(leaky(el[(size_t)src[e] * GHEADS + tid] + erd) - m);
        s_m[tid] = m;
        s_inv[tid] = s > 0.f ? 1.0f / s : 0.f;
    }
    __syncthreads();

    float acc = 0.f;
    for (int base = e0; base < e1; base += 64) {
        int cnt = min(64, e1 - base);
        if (tid < cnt) s_src[tid] = src[base + tid];
        __syncthreads();
        // alpha for (edge, head) computed once into LDS
        if (tid < cnt * GHEADS) {
            int le = tid >> 2, lh = tid & 3;
            float x = leaky(el[(size_t)s_src[le] * GHEADS + lh] + s_er[lh]);
            s_alpha[le * GHEADS + lh] = __expf(x - s_m[lh]) * s_inv[lh];
        }
        __syncthreads();
        for (int le = 0; le < cnt; ++le) {
            acc += s_alpha[le * GHEADS + hd] *
                   h[(size_t)s_src[le] * GD + hd * GHID + o];
        }
        __syncthreads();
    }

    acc += bias[hd * GHID + o];

    if (!MEAN) {
        out[(size_t)d * GD + tid] = acc;
    } else {
        s_red[tid] = acc;
        __syncthreads();
        if (tid < 64) {
            float v = (s_red[tid] + s_red[tid + 64] +
                       s_red[tid + 128] + s_red[tid + 192]) * 0.25f;
            out[(size_t)d * 64 + tid] = v;
        }
    }
}

// ---------------------------------------------------------------------------
extern "C" void kernel_launch(void* const* d_in, const int* in_sizes, int n_in,
                              void* d_out, int out_size, void* d_ws, size_t ws_size,
                              hipStream_t stream) {
    const float* x   = (const float*)d_in[0];
    const int*   src = (const int*)d_in[1];
    const int*   dst = (const int*)d_in[2];
    const float* W1  = (const float*)d_in[3];
    const float* al1 = (const float*)d_in[4];
    const float* ar1 = (const float*)d_in[5];
    const float* b1  = (const float*)d_in[6];
    const float* W2  = (const float*)d_in[7];
    const float* al2 = (const float*)d_in[8];
    const float* ar2 = (const float*)d_in[9];
    const float* b2  = (const float*)d_in[10];
    float* out = (float*)d_out;

    char* ws = (char*)d_ws;
    size_t off = 0;
    auto alloc = [&](size_t bytes) {
        size_t o = off;
        off += (bytes + 255) & ~(size_t)255;
        return o;
    };
    int*   rowptr = (int*)(ws + alloc((GN + 1) * sizeof(int)));
    float* hraw   = (float*)(ws + alloc((size_t)GN * GD * sizeof(float))); // reused layer2
    float* hout   = (float*)(ws + alloc((size_t)GN * GD * sizeof(float)));
    float* el     = (float*)(ws + alloc((size_t)GN * GHEADS * sizeof(float)));
    float* er     = (float*)(ws + alloc((size_t)GN * GHEADS * sizeof(float)));
    (void)ws_size; (void)in_sizes; (void)n_in; (void)out_size;

    // CSR from sorted dst (shared by both layers)
    build_rowptr<<<(GN + 1 + 255) / 256, 256, 0, stream>>>(dst, rowptr, GE, GN);

    // ---- Layer 1 ----
    gemm_wmma_f32<<<dim3(GN / 16, 2), 256, 0, stream>>>(x, W1, hraw);
    el_er_kernel<<<(GN * GHEADS + 255) / 256, 256, 0, stream>>>(hraw, al1, ar1, el, er);
    gat_aggregate<false><<<GN, 256, 0, stream>>>(hraw, el, er, src, rowptr, b1, hout);

    // ---- Layer 2 (fused head-mean into epilogue) ----
    gemm_wmma_f32<<<dim3(GN / 16, 2), 256, 0, stream>>>(hout, W2, hraw);
    el_er_kernel<<<(GN * GHEADS + 255) / 256, 256, 0, stream>>>(hraw, al2, ar2, el, er);
    gat_aggregate<true><<<GN, 256, 0, stream>>>(hraw, el, er, src, rowptr, b2, out);
}